// BaselineDNN_43782896615919
// MI455X (gfx1250) — compile-verified
//
#include <hip/hip_runtime.h>
#include <hip/hip_bf16.h>

typedef __attribute__((ext_vector_type(2))) float v2f;
typedef __attribute__((ext_vector_type(8))) float v8f;

#define B_ROWS   4096
#define MAX_LEN  128
#define EMB_DIM  300
#define REP_DIM  600
#define HIDDEN_N 1000
#define NCLASS   5
#define NT       4          // 16-wide N tiles per wave in GEMM1 (16x64 per wave)

// ---------------------------------------------------------------------------
// Kernel 1: embedding gather + masked avg/max pool.
// One block per batch row, 128 threads striding the 300 dims (coalesced).
// Prefetch next token's row (global_prefetch_b8) to hide gather latency.
// ---------------------------------------------------------------------------
__global__ void pool_kernel(const int* __restrict__ x,
                            const int* __restrict__ lengths,
                            const float* __restrict__ emb,
                            float* __restrict__ rep) {
    const int b   = blockIdx.x;
    const int tid = threadIdx.x;          // 0..127
    int len = lengths[b];
    if (len < 1) len = 1;

    const int* xrow = x + b * MAX_LEN;

    float s0 = 0.f, s1 = 0.f, s2 = 0.f;
    const float NEG = -3.402823466e38f;   // finfo(f32).min
    float m0 = NEG, m1 = NEG, m2 = NEG;

    for (int t = 0; t < len; ++t) {
        const float* e = emb + (long long)xrow[t] * EMB_DIM;
        if (t + 1 < len) {
            // overlap next row's L2 access with this row's accumulation
            const float* enext = emb + (long long)xrow[t + 1] * EMB_DIM;
            __builtin_prefetch(enext + tid, 0, 3);
            __builtin_prefetch(enext + tid + 128, 0, 3);
        }
        float v0 = e[tid];
        float v1 = e[tid + 128];
        s0 += v0;  m0 = fmaxf(m0, v0);
        s1 += v1;  m1 = fmaxf(m1, v1);
        if (tid < EMB_DIM - 256) {        // dims 256..299
            float v2 = e[tid + 256];
            s2 += v2;  m2 = fmaxf(m2, v2);
        }
    }

    const float inv = 1.0f / (float)len;
    float* r = rep + (long long)b * REP_DIM;
    r[tid]                 = s0 * inv;
    r[tid + 128]           = s1 * inv;
    r[EMB_DIM + tid]       = m0;
    r[EMB_DIM + tid + 128] = m1;
    if (tid < EMB_DIM - 256) {
        r[tid + 256]           = s2 * inv;
        r[EMB_DIM + tid + 256] = m2;
    }
}

// ---------------------------------------------------------------------------
// Kernel 2: hidden = relu(rep @ W1 + b1) via V_WMMA_F32_16X16X4_F32.
// One wave per 16x64 output tile: the A fragment (one global_load_b64 per
// k-step) is reused across NT=4 WMMAs, cutting rep re-reads 4x.
// A 16x4 layout: lanes 0-15 hold K=k0..k0+1, lanes 16-31 hold K=k0+2..k0+3,
// M = lane & 15. B 4x16 mirrored with N = lane & 15.
// ---------------------------------------------------------------------------
__global__ void gemm1_kernel(const float* __restrict__ rep,
                             const float* __restrict__ W1,
                             const float* __restrict__ b1,
                             float* __restrict__ hidden) {
    const int m0   = blockIdx.x * 16;
    const int n0   = blockIdx.y * (16 * NT);
    const int lane = threadIdx.x;              // 0..31, full wave, EXEC all-ones
    const int mrow  = lane & 15;
    const int khalf = (lane >> 4) * 2;         // 0 or 2
    const int ncol  = lane & 15;

    int n[NT], nc[NT];
    #pragma unroll
    for (int j = 0; j < NT; ++j) {
        n[j]  = n0 + 16 * j + ncol;
        nc[j] = (n[j] < HIDDEN_N) ? n[j] : (HIDDEN_N - 1);   // clamp loads, no OOB
    }

    const float* arow = rep + (long long)(m0 + mrow) * REP_DIM;

    v8f c[NT];
    #pragma unroll
    for (int j = 0; j < NT; ++j) c[j] = (v8f){};

    for (int k0 = 0; k0 < REP_DIM; k0 += 4) {
        v2f a;
        a.x = arow[k0 + khalf];
        a.y = arow[k0 + khalf + 1];
        const float* wrow0 = W1 + (long long)(k0 + khalf) * HIDDEN_N;
        const float* wrow1 = wrow0 + HIDDEN_N;
        #pragma unroll
        for (int j = 0; j < NT; ++j) {
            v2f bm;
            bm.x = wrow0[nc[j]];
            bm.y = wrow1[nc[j]];
            // 8 args: (neg_a, A, neg_b, B, c_mod, C, reuse_a, reuse_b)
            c[j] = __builtin_amdgcn_wmma_f32_16x16x4_f32(
                       false, a, false, bm, (short)0, c[j], false, false);
        }
    }

    const int mbase = m0 + ((lane >> 4) * 8);   // C layout: M = r + 8*(lane>=16)
    #pragma unroll
    for (int j = 0; j < NT; ++j) {
        if (n[j] < HIDDEN_N) {
            const float bias = b1[nc[j]];
            #pragma unroll
            for (int r = 0; r < 8; ++r) {
                float v = c[j][r] + bias;
                v = v > 0.f ? v : 0.f;
                hidden[(long long)(mbase + r) * HIDDEN_N + n[j]] = v;
            }
        }
    }
}

// ---------------------------------------------------------------------------
// Kernel 3: logits = hidden @ W2 + b2.  N=5 is too skinny for WMMA; one wave
// per batch row with 5 register accumulators + wave32 shuffle reduction.
// ---------------------------------------------------------------------------
__global__ void gemm2_kernel(const float* __restrict__ hidden,
                             const float* __restrict__ W2,
                             const float* __restrict__ b2,
                             float* __restrict__ out) {
    const int b    = blockIdx.x;
    const int lane = threadIdx.x;              // 0..31
    const float* h = hidden + (long long)b * HIDDEN_N;

    float acc[NCLASS] = {0.f, 0.f, 0.f, 0.f, 0.f};
    for (int k = lane; k < HIDDEN_N; k += 32) {
        const float hv = h[k];
        const float* w = W2 + (long long)k * NCLASS;
        #pragma unroll
        for (int c = 0; c < NCLASS; ++c) acc[c] += hv * w[c];
    }

    #pragma unroll
    for (int off = 16; off > 0; off >>= 1) {
        #pragma unroll
        for (int c = 0; c < NCLASS; ++c)
            acc[c] += __shfl_down(acc[c], off, 32);
    }

    if (lane == 0) {
        #pragma unroll
        for (int c = 0; c < NCLASS; ++c)
            out[(long long)b * NCLASS + c] = acc[c] + b2[c];
    }
}

// ---------------------------------------------------------------------------
extern "C" void kernel_launch(void* const* d_in, const int* in_sizes, int n_in,
                              void* d_out, int out_size, void* d_ws, size_t ws_size,
                              hipStream_t stream) {
    const int*   x       = (const int*)d_in[0];
    const int*   lengths = (const int*)d_in[1];
    const float* emb     = (const float*)d_in[2];
    const float* W1      = (const float*)d_in[3];
    const float* b1      = (const float*)d_in[4];
    const float* W2      = (const float*)d_in[5];
    const float* b2      = (const float*)d_in[6];
    float*       out     = (float*)d_out;

    float* rep    = (float*)d_ws;                              // 4096*600 f32
    float* hidden = rep + (size_t)B_ROWS * REP_DIM;            // 4096*1000 f32

    pool_kernel<<<B_ROWS, 128, 0, stream>>>(x, lengths, emb, rep);

    dim3 g1(B_ROWS / 16, (HIDDEN_N + 16 * NT - 1) / (16 * NT));
    gemm1_kernel<<<g1, 32, 0, stream>>>(rep, W1, b1, hidden);

    gemm2_kernel<<<B_ROWS, 32, 0, stream>>>(hidden, W2, b2, out);
}